// YOLOv1Loss_13606456394545
// MI455X (gfx1250) — compile-verified
//
#include <hip/hip_runtime.h>
#include <stdint.h>

// ---------------------------------------------------------------------------
// YOLOv1 loss on MI455X (gfx1250).
// Memory-bound problem (~96 MB streamed once -> ~4.1 us at 23.3 TB/s).
// Data movement uses the CDNA5 Tensor Data Mover (tensor_load_to_lds) with
// double buffering tracked by TENSORcnt; all math runs out of LDS in fp32.
// No matmul structure => WMMA intentionally not used.
// ---------------------------------------------------------------------------

typedef uint32_t u32x4 __attribute__((ext_vector_type(4)));
typedef uint32_t u32x8 __attribute__((ext_vector_type(8)));

#define CELLS   49
#define CH      30
#define PERB    (CELLS * CH)   // 1470 floats per batch item per tensor
#define NBOX    98
#define NBLK    2048
#define NTHR    128

// Issue one TDM 1-D tile load: ndw fp32 elements, global -> LDS.
// D# layout per CDNA5 ISA ch.8:
//   group0: [0]=count=1, [1]=lds byte addr, [2]=gaddr[31:0],
//           [3]=gaddr[56:32] | type(2)<<30
//   group1: [0]=data_size(4B)=2<<16, [1]=tensor_dim0[15:0]<<16,
//           [2]=tensor_dim0[31:16] | tensor_dim1(=1)<<16,
//           [3]=tile_dim0<<16, [4]=tile_dim1(=1), [5]=tensor_dim0_stride[31:0]
//   group2: [0]=tensor_dim2(=1); group3: zeros
__device__ __forceinline__ void tdm_load_1d(const float* gsrc, float* ldsdst,
                                            uint32_t ndw) {
  uint64_t ga = (uint64_t)(uintptr_t)gsrc;
  uint32_t la = (uint32_t)(uintptr_t)(void*)ldsdst;  // flat shared: low 32b = LDS offset
  u32x4 g0 = { 1u, la, (uint32_t)ga,
               (uint32_t)((ga >> 32) & 0x1FFFFFFu) | (2u << 30) };
  u32x8 g1 = { (2u << 16),
               (ndw & 0xFFFFu) << 16,
               (ndw >> 16) | (1u << 16),
               (ndw & 0xFFFFu) << 16,
               1u,
               ndw,
               0u, 0u };
  u32x4 g2 = { 1u, 0u, 0u, 0u };
  u32x4 g3 = { 0u, 0u, 0u, 0u };
  asm volatile("tensor_load_to_lds %0, %1, %2, %3"
               :: "s"(g0), "s"(g1), "s"(g2), "s"(g3)
               : "memory");
}

__global__ __launch_bounds__(NTHR) void yolo_loss_main(
    const float* __restrict__ pred, const float* __restrict__ targ,
    float* __restrict__ ws, int B) {
  __shared__ float sbuf[2][2][PERB];           // [slot][pred/targ][1470]
  __shared__ float bx1[NBOX], by1[NBOX], bx2[NBOX], by2[NBOX];
  __shared__ float bar[NBOX], bmsk[NBOX];
  __shared__ float red[NTHR];

  const int tid = threadIdx.x;
  const int gsz = gridDim.x;
  const int wid = __builtin_amdgcn_readfirstlane(tid >> 5);  // wave id (uniform)

  float l_cls = 0.f, l_objc = 0.f, l_coord = 0.f;

  int bi0 = blockIdx.x;
  if (bi0 < B && wid == 0) {                   // prologue DMA (TDM ignores EXEC;
    tdm_load_1d(pred + (size_t)bi0 * PERB, sbuf[0][0], PERB);   // wave-level issue)
    tdm_load_1d(targ + (size_t)bi0 * PERB, sbuf[0][1], PERB);
  }

  int s = 0;
  for (int bi = bi0; bi < B; bi += gsz, s ^= 1) {
    int nbi = bi + gsz;
    if (wid == 0) {
      if (nbi < B) {                           // prefetch next tile into other slot
        tdm_load_1d(pred + (size_t)nbi * PERB, sbuf[s ^ 1][0], PERB);
        tdm_load_1d(targ + (size_t)nbi * PERB, sbuf[s ^ 1][1], PERB);
        __builtin_amdgcn_s_wait_tensorcnt(2);  // current pair done; next may fly
      } else {
        __builtin_amdgcn_s_wait_tensorcnt(0);
      }
    }
    __syncthreads();                           // publish LDS tile to all waves

    const float* P = sbuf[s][0];
    const float* T = sbuf[s][1];

    // ---- Phase A: cache pred boxes (xyxy, area) + object mask in LDS ----
    if (tid < NBOX) {
      int c = tid >> 1;
      const float* pb = P + c * CH + (tid & 1) * 5;
      float cx = pb[0], cy = pb[1], w = pb[2], h = pb[3];
      float x1 = cx - 0.5f * w, y1 = cy - 0.5f * h;
      float x2 = cx + 0.5f * w, y2 = cy + 0.5f * h;
      bx1[tid] = x1; by1[tid] = y1; bx2[tid] = x2; by2[tid] = y2;
      bar[tid] = (x2 - x1) * (y2 - y1);
      bmsk[tid] = (T[c * CH + 4] > 0.f) ? 1.f : 0.f;
    }
    __syncthreads();

    // ---- Phase B1: per-cell noobj + class losses (threads 0..48) ----
    if (tid < CELLS) {
      const float* pc = P + tid * CH;
      const float* tc = T + tid * CH;
      float ct = tc[4];
      if (ct == 0.f) {
        float d4 = pc[4] - ct;
        float d9 = pc[9] - tc[9];
        l_objc += 0.5f * (d4 * d4 + d9 * d9);  // NOOBJ_LAMBDA folded in
      } else {
        float acc = 0.f;
#pragma unroll
        for (int k = 10; k < 30; ++k) { float d = pc[k] - tc[k]; acc += d * d; }
        l_cls += acc;
      }
    }

    // ---- Phase B2: per-target-box masked IoU max + coord/obj losses ----
    if (tid < NBOX && bmsk[tid] != 0.f) {
      int c = tid >> 1;
      int off = c * CH + (tid & 1) * 5;
      const float* tb = T + off;
      float tcx = tb[0], tcy = tb[1], tw = tb[2], th2 = tb[3], tcf = tb[4];
      float tx1 = tcx - 0.5f * tw, ty1 = tcy - 0.5f * th2;
      float tx2 = tcx + 0.5f * tw, ty2 = tcy + 0.5f * th2;
      float tA = (tx2 - tx1) * (ty2 - ty1);
      float mx = 0.f;
      for (int n = 0; n < NBOX; ++n) {
        if (bmsk[n] != 0.f) {                  // block-uniform -> no divergence
          float lx = fmaxf(bx1[n], tx1), ly = fmaxf(by1[n], ty1);
          float rx = fminf(bx2[n], tx2), ry = fminf(by2[n], ty2);
          float iw = fmaxf(rx - lx, 0.f), ih = fmaxf(ry - ly, 0.f);
          float inter = iw * ih;
          float uni = bar[n] + tA - inter;
          float iou = inter / (uni > 0.f ? uni : 1.f);
          mx = fmaxf(mx, iou);
        }
      }
      if (mx != 0.f) {                         // cmask
        const float* pb = P + off;
        float dcx = pb[0] - tcx, dcy = pb[1] - tcy;
        float dw = sqrtf(pb[2]) - sqrtf(tw);
        float dh = sqrtf(pb[3]) - sqrtf(th2);
        l_coord += dcx * dcx + dcy * dcy + dw * dw + dh * dh;
        float dc = pb[4] - tcf;
        l_objc += dc * dc;
      }
    }
    __syncthreads();                           // all reads of slot s done before
                                               // next iteration's DMA reuses it
  }

  // ---- deterministic block reduction of the 3 partials ----
  float vals[3] = { l_cls, l_objc, l_coord };
#pragma unroll
  for (int j = 0; j < 3; ++j) {
    __syncthreads();
    red[tid] = vals[j];
    __syncthreads();
    for (int st = NTHR / 2; st > 0; st >>= 1) {
      if (tid < st) red[tid] += red[tid + st];
      __syncthreads();
    }
    if (tid == 0) ws[(size_t)blockIdx.x * 3 + j] = red[0];
  }
}

__global__ __launch_bounds__(256) void yolo_loss_reduce(
    const float* __restrict__ ws, int nblk, float* __restrict__ out, float invB) {
  __shared__ double red0[256], red1[256], red2[256];
  int tid = threadIdx.x;
  double s0 = 0.0, s1 = 0.0, s2 = 0.0;
  for (int i = tid; i < nblk; i += 256) {
    s0 += (double)ws[(size_t)i * 3 + 0];
    s1 += (double)ws[(size_t)i * 3 + 1];
    s2 += (double)ws[(size_t)i * 3 + 2];
  }
  red0[tid] = s0; red1[tid] = s1; red2[tid] = s2;
  __syncthreads();
  for (int st = 128; st > 0; st >>= 1) {
    if (tid < st) {
      red0[tid] += red0[tid + st];
      red1[tid] += red1[tid + st];
      red2[tid] += red2[tid + st];
    }
    __syncthreads();
  }
  if (tid == 0) {
    float cls   = (float)red0[0] * invB;
    float objc  = (float)red1[0] * invB;
    float coord = (float)red2[0] * 5.0f * invB;   // COORD_LAMBDA
    out[0] = cls + objc + coord;
    out[1] = cls;
    out[2] = objc;
    out[3] = coord;
  }
}

extern "C" void kernel_launch(void* const* d_in, const int* in_sizes, int n_in,
                              void* d_out, int out_size, void* d_ws, size_t ws_size,
                              hipStream_t stream) {
  (void)n_in; (void)out_size; (void)ws_size;
  const float* pred = (const float*)d_in[0];
  const float* targ = (const float*)d_in[1];
  int B = in_sizes[0] / PERB;                 // 8192 for the reference shapes
  float* ws = (float*)d_ws;                   // NBLK*3 floats = 24 KB partials
  int nblk = NBLK < B ? NBLK : B;
  yolo_loss_main<<<nblk, NTHR, 0, stream>>>(pred, targ, ws, B);
  yolo_loss_reduce<<<1, 256, 0, stream>>>(ws, nblk, (float*)d_out, 1.0f / (float)B);
}